// GAT_77008763617447
// MI455X (gfx1250) — compile-verified
//
#include <hip/hip_runtime.h>
#include <math.h>

#define NNODES 50000
#define NEDGES 800000
#define IN_DIM 128
#define HID 128
#define HEADS 8
#define C1 16
#define OUT_DIM 64
#define NEG_SLOPE 0.2f

typedef float v2f __attribute__((ext_vector_type(2)));
typedef float v8f __attribute__((ext_vector_type(8)));

// ---------------- utility kernels ----------------

__global__ void fill_u32_kernel(unsigned int* __restrict__ p, long long n, unsigned int v) {
    long long i = (long long)blockIdx.x * blockDim.x + threadIdx.x;
    if (i < n) p[i] = v;
}

// Monotone float<->uint order encoding for atomicMax-based segment max.
__device__ __forceinline__ unsigned int f32_order_encode(float f) {
    unsigned int b = __float_as_uint(f);
    return (b & 0x80000000u) ? ~b : (b | 0x80000000u);
}
__device__ __forceinline__ float f32_order_decode(unsigned int k) {
    unsigned int b = (k & 0x80000000u) ? (k ^ 0x80000000u) : ~k;
    return __uint_as_float(b);
}

__global__ void decode_max_kernel(unsigned int* __restrict__ m, long long n) {
    long long i = (long long)blockIdx.x * blockDim.x + threadIdx.x;
    if (i < n) {
        float f = f32_order_decode(m[i]);
        m[i] = __float_as_uint(f);
    }
}

__device__ __forceinline__ float lrelu(float x) {
    return x >= 0.0f ? x : NEG_SLOPE * x;
}

// ---------------- WMMA GEMM: C[rows,cols] = A[rows,K] @ B[K,cols] ----------------
// One wave computes a 16x16 tile, K-loop step 4 using V_WMMA_F32_16X16X4_F32.
// rows % 16 == 0, cols % 16 == 0, K % 4 == 0 (all exact here).
__global__ void gemm_wmma_f32(const float* __restrict__ A, const float* __restrict__ B,
                              float* __restrict__ C, int rows, int K, int cols) {
    const int lane = threadIdx.x & 31;
    const int wave = threadIdx.x >> 5;
    const int wavesPerBlock = blockDim.x >> 5;
    const int tile = blockIdx.x * wavesPerBlock + wave;
    const int colTiles = cols >> 4;
    const int rowTile = tile / colTiles;
    const int colTile = tile - rowTile * colTiles;
    if (rowTile * 16 >= rows) return;

    const int half = lane >> 4;   // 0: K+0/K+1, 1: K+2/K+3
    const int l    = lane & 15;

    const float* __restrict__ arow = A + (size_t)(rowTile * 16 + l) * K;
    const int col = colTile * 16 + l;
    const int kOff = half * 2;

    v8f c = {0.f, 0.f, 0.f, 0.f, 0.f, 0.f, 0.f, 0.f};
    for (int k = 0; k < K; k += 4) {
        v2f a, b;
        a.x = arow[k + kOff + 0];
        a.y = arow[k + kOff + 1];
        b.x = B[(size_t)(k + kOff + 0) * cols + col];
        b.y = B[(size_t)(k + kOff + 1) * cols + col];
        c = __builtin_amdgcn_wmma_f32_16x16x4_f32(false, a, false, b, (short)0, c,
                                                  false, false);
    }
    // C/D layout: VGPR v holds row M = v + 8*half, column = lane&15.
    #pragma unroll
    for (int v = 0; v < 8; ++v) {
        int row = rowTile * 16 + v + half * 8;
        C[(size_t)row * cols + col] = c[v];
    }
}

// ---------------- per-node attention terms ----------------
// alphaS[n,h] = sum_c Hf[n, h*Ch + c] * a_s[h,c]   (same for alphaD)
__global__ void alpha_kernel(const float* __restrict__ Hf,
                             const float* __restrict__ a_s, const float* __restrict__ a_d,
                             float* __restrict__ aS, float* __restrict__ aD,
                             int n, int heads, int chead) {
    long long t = (long long)blockIdx.x * blockDim.x + threadIdx.x;
    if (t >= (long long)n * heads) return;
    int node = (int)(t / heads);
    int h = (int)(t - (long long)node * heads);
    const float* hr = Hf + (size_t)node * heads * chead + h * chead;
    const float* as = a_s + h * chead;
    const float* ad = a_d + h * chead;
    float ss = 0.f, sd = 0.f;
    for (int c = 0; c < chead; ++c) {
        float v = hr[c];
        ss += v * as[c];
        sd += v * ad[c];
    }
    aS[t] = ss;
    aD[t] = sd;
}

// ---------------- edge pass A: segment max (order-encoded atomicMax) ----------------
__global__ void edge_max_kernel(const int* __restrict__ ei, int E, int n,
                                const float* __restrict__ aS, const float* __restrict__ aD,
                                unsigned int* __restrict__ mEnc, int H) {
    long long t = (long long)blockIdx.x * blockDim.x + threadIdx.x;
    long long Etot = (long long)E + n;
    if (t >= Etot * H) return;
    int e = (int)(t / H);
    int h = (int)(t - (long long)e * H);
    int s = (e < E) ? ei[e]     : (e - E);
    int d = (e < E) ? ei[E + e] : (e - E);
    float sc = lrelu(aS[(size_t)s * H + h] + aD[(size_t)d * H + h]);
    atomicMax(&mEnc[(size_t)d * H + h], f32_order_encode(sc));
}

// ---------------- edge pass B: segment sum of exp(score - max) ----------------
__global__ void edge_sum_kernel(const int* __restrict__ ei, int E, int n,
                                const float* __restrict__ aS, const float* __restrict__ aD,
                                const float* __restrict__ m, float* __restrict__ denom,
                                int H) {
    long long t = (long long)blockIdx.x * blockDim.x + threadIdx.x;
    long long Etot = (long long)E + n;
    if (t >= Etot * H) return;
    int e = (int)(t / H);
    int h = (int)(t - (long long)e * H);
    int s = (e < E) ? ei[e]     : (e - E);
    int d = (e < E) ? ei[E + e] : (e - E);
    float sc = lrelu(aS[(size_t)s * H + h] + aD[(size_t)d * H + h]);
    float ex = expf(sc - m[(size_t)d * H + h]);
    atomicAdd(&denom[(size_t)d * H + h], ex);
}

// ---------------- edge pass C: weighted scatter-add aggregation ----------------
// F = H*C features per node; F consecutive lanes cover one edge -> coalesced.
__global__ void edge_agg_kernel(const int* __restrict__ ei, int E, int n,
                                const float* __restrict__ aS, const float* __restrict__ aD,
                                const float* __restrict__ m, const float* __restrict__ denom,
                                const float* __restrict__ Hf, float* __restrict__ agg,
                                int H, int C) {
    const int F = H * C;
    long long t = (long long)blockIdx.x * blockDim.x + threadIdx.x;
    long long Etot = (long long)E + n;
    if (t >= Etot * F) return;
    int e = (int)(t / F);
    int r = (int)(t - (long long)e * F);
    int h = r / C;
    int s = (e < E) ? ei[e]     : (e - E);
    int d = (e < E) ? ei[E + e] : (e - E);
    float sc = lrelu(aS[(size_t)s * H + h] + aD[(size_t)d * H + h]);
    float alpha = expf(sc - m[(size_t)d * H + h]) / denom[(size_t)d * H + h];
    atomicAdd(&agg[(size_t)d * F + r], alpha * Hf[(size_t)s * F + r]);
}

// ---------------- layer-1 epilogue: x2 = elu(agg + b) (in place) ----------------
__global__ void elu_bias_kernel(float* __restrict__ agg, const float* __restrict__ b,
                                long long total, int F) {
    long long i = (long long)blockIdx.x * blockDim.x + threadIdx.x;
    if (i >= total) return;
    float v = agg[i] + b[(int)(i % F)];
    agg[i] = v > 0.0f ? v : expm1f(v);
}

// ---------------- final: out = log_softmax(agg2 + b2), one wave32 per node ----------------
__global__ void logsoftmax_kernel(const float* __restrict__ agg, const float* __restrict__ b,
                                  float* __restrict__ out, int n) {
    int wave = threadIdx.x >> 5;
    int lane = threadIdx.x & 31;
    int node = blockIdx.x * (blockDim.x >> 5) + wave;
    if (node >= n) return;
    const float* row = agg + (size_t)node * OUT_DIM;
    float v0 = row[lane]      + b[lane];
    float v1 = row[lane + 32] + b[lane + 32];
    float mx = fmaxf(v0, v1);
    #pragma unroll
    for (int off = 16; off > 0; off >>= 1)
        mx = fmaxf(mx, __shfl_xor(mx, off, 32));
    float s = expf(v0 - mx) + expf(v1 - mx);
    #pragma unroll
    for (int off = 16; off > 0; off >>= 1)
        s += __shfl_xor(s, off, 32);
    float lse = mx + logf(s);
    float* orow = out + (size_t)node * OUT_DIM;
    orow[lane]      = v0 - lse;
    orow[lane + 32] = v1 - lse;
}

// ---------------- host orchestration ----------------

static inline size_t align256(size_t x) { return (x + 255) & ~(size_t)255; }

extern "C" void kernel_launch(void* const* d_in, const int* in_sizes, int n_in,
                              void* d_out, int out_size, void* d_ws, size_t ws_size,
                              hipStream_t stream) {
    const float* x      = (const float*)d_in[0];
    const int*   ei     = (const int*)d_in[1];     // [2, E] int32
    const float* W1     = (const float*)d_in[2];
    const float* a_src1 = (const float*)d_in[3];
    const float* a_dst1 = (const float*)d_in[4];
    const float* b1     = (const float*)d_in[5];
    const float* W2     = (const float*)d_in[6];
    const float* a_src2 = (const float*)d_in[7];
    const float* a_dst2 = (const float*)d_in[8];
    const float* b2     = (const float*)d_in[9];
    float* out = (float*)d_out;

    const int N = NNODES, E = NEDGES;
    const long long Etot = (long long)E + N;

    // workspace carve-up
    char* w = (char*)d_ws;
    float* H1    = (float*)w; w += align256((size_t)N * HID * 4);
    float* X2    = (float*)w; w += align256((size_t)N * HID * 4);      // agg1, then elu'd in place
    float* H2    = (float*)w; w += align256((size_t)N * OUT_DIM * 4);
    float* AGG2  = (float*)w; w += align256((size_t)N * OUT_DIM * 4);
    float* aS1   = (float*)w; w += align256((size_t)N * HEADS * 4);
    float* aD1   = (float*)w; w += align256((size_t)N * HEADS * 4);
    float* m1    = (float*)w; w += align256((size_t)N * HEADS * 4);
    float* den1  = (float*)w; w += align256((size_t)N * HEADS * 4);
    float* aS2   = (float*)w; w += align256((size_t)N * 4);
    float* aD2   = (float*)w; w += align256((size_t)N * 4);
    float* m2    = (float*)w; w += align256((size_t)N * 4);
    float* den2  = (float*)w; w += align256((size_t)N * 4);
    (void)ws_size; (void)n_in; (void)in_sizes; (void)out_size;

    const unsigned int NEG_INF_ENC = 0x007FFFFFu;  // f32_order_encode(-inf)
    const int T = 256;
    auto blocks = [](long long cnt, int t) { return (unsigned int)((cnt + t - 1) / t); };

    // ---------- Layer 1 ----------
    fill_u32_kernel<<<blocks((long long)N * HEADS, T), T, 0, stream>>>((unsigned int*)m1, (long long)N * HEADS, NEG_INF_ENC);
    fill_u32_kernel<<<blocks((long long)N * HEADS, T), T, 0, stream>>>((unsigned int*)den1, (long long)N * HEADS, 0u);
    fill_u32_kernel<<<blocks((long long)N * HID, T), T, 0, stream>>>((unsigned int*)X2, (long long)N * HID, 0u);

    // H1 = x @ W1 : tiles = (N/16)*(HID/16) = 3125*8 = 25000 waves, 4 waves/block
    gemm_wmma_f32<<<25000 / 4, 128, 0, stream>>>(x, W1, H1, N, IN_DIM, HID);

    alpha_kernel<<<blocks((long long)N * HEADS, T), T, 0, stream>>>(H1, a_src1, a_dst1, aS1, aD1, N, HEADS, C1);

    edge_max_kernel<<<blocks(Etot * HEADS, T), T, 0, stream>>>(ei, E, N, aS1, aD1, (unsigned int*)m1, HEADS);
    decode_max_kernel<<<blocks((long long)N * HEADS, T), T, 0, stream>>>((unsigned int*)m1, (long long)N * HEADS);
    edge_sum_kernel<<<blocks(Etot * HEADS, T), T, 0, stream>>>(ei, E, N, aS1, aD1, m1, den1, HEADS);
    edge_agg_kernel<<<blocks(Etot * HID, T), T, 0, stream>>>(ei, E, N, aS1, aD1, m1, den1, H1, X2, HEADS, C1);

    elu_bias_kernel<<<blocks((long long)N * HID, T), T, 0, stream>>>(X2, b1, (long long)N * HID, HID);

    // ---------- Layer 2 ----------
    fill_u32_kernel<<<blocks((long long)N, T), T, 0, stream>>>((unsigned int*)m2, (long long)N, NEG_INF_ENC);
    fill_u32_kernel<<<blocks((long long)N, T), T, 0, stream>>>((unsigned int*)den2, (long long)N, 0u);
    fill_u32_kernel<<<blocks((long long)N * OUT_DIM, T), T, 0, stream>>>((unsigned int*)AGG2, (long long)N * OUT_DIM, 0u);

    // H2 = X2 @ W2 : tiles = 3125*4 = 12500 waves
    gemm_wmma_f32<<<12500 / 4, 128, 0, stream>>>(X2, W2, H2, N, HID, OUT_DIM);

    alpha_kernel<<<blocks((long long)N, T), T, 0, stream>>>(H2, a_src2, a_dst2, aS2, aD2, N, 1, OUT_DIM);

    edge_max_kernel<<<blocks(Etot, T), T, 0, stream>>>(ei, E, N, aS2, aD2, (unsigned int*)m2, 1);
    decode_max_kernel<<<blocks((long long)N, T), T, 0, stream>>>((unsigned int*)m2, (long long)N);
    edge_sum_kernel<<<blocks(Etot, T), T, 0, stream>>>(ei, E, N, aS2, aD2, m2, den2, 1);
    edge_agg_kernel<<<blocks(Etot * OUT_DIM, T), T, 0, stream>>>(ei, E, N, aS2, aD2, m2, den2, H2, AGG2, 1, OUT_DIM);

    // out = log_softmax(AGG2 + b2); 8 wave32 per block
    logsoftmax_kernel<<<blocks((long long)N, 8), 256, 0, stream>>>(AGG2, b2, out, N);
}